// GraphConvolution_16758962389075
// MI455X (gfx1250) — compile-verified
//
#include <hip/hip_runtime.h>
#include <hip/hip_bf16.h>

#define D_IN 128
#define D_OUT 64
#define BN_EPS 1e-3f

typedef __attribute__((ext_vector_type(2))) float v2f;
typedef __attribute__((ext_vector_type(8))) float v8f;

// ---------------------------------------------------------------------------
// K0: zero output accumulator and BN stats scratch
// ---------------------------------------------------------------------------
__global__ __launch_bounds__(256) void gc_zero_kernel(float* __restrict__ out, int nOut,
                                                      float* __restrict__ stats) {
    int i = blockIdx.x * 256 + threadIdx.x;
    if (i < nOut) out[i] = 0.0f;
    if (i < 2 * D_OUT) stats[i] = 0.0f;
}

// ---------------------------------------------------------------------------
// K1: pre_sup = X @ W via V_WMMA_F32_16X16X4_F32.
// 256 threads = 8 waves; each wave computes a 16x64 output tile.
// W staged in LDS *pair-swizzled*: sWp[(k>>1)*64 + n] = {W[k,n], W[k+1,n]}
// so each lane's B fragment is one ds_load_b64 (no register shuffles).
// ---------------------------------------------------------------------------
__global__ __launch_bounds__(256) void gc_gemm_kernel(const float* __restrict__ X,
                                                      const float* __restrict__ W,
                                                      float* __restrict__ P,
                                                      int nRows) {
    __shared__ v2f sWp[(D_IN / 2) * D_OUT];  // 64 k-pairs x 64 cols = 32 KB
    for (int idx = threadIdx.x; idx < D_IN * D_OUT; idx += 256) {
        int k = idx >> 6;        // 0..127
        int n = idx & 63;        // 0..63
        sWp[(k >> 1) * D_OUT + n][k & 1] = W[idx];
    }
    __syncthreads();

    const int lane  = threadIdx.x & 31;
    const int wave  = threadIdx.x >> 5;
    const int tile  = blockIdx.x * 8 + wave;   // 16-row tile index
    const int row0  = tile * 16;
    if (row0 >= nRows) return;

    const int m     = lane & 15;               // M row (A) / N col (B,C)
    const int khalf = (lane >> 4) << 1;        // 0 for lanes 0-15, 2 for 16-31

    // clamp row for safe tail loads (tail rows are never stored)
    int arow = row0 + m;
    if (arow >= nRows) arow = nRows - 1;
    const float* __restrict__ xrow = X + (size_t)arow * D_IN;

    v8f acc0 = {}, acc1 = {}, acc2 = {}, acc3 = {};

    for (int k = 0; k < D_IN; k += 4) {
        // A fragment 16x4: VGPR0 = K=k+khalf, VGPR1 = K=k+khalf+1
        v2f a;
        a.x = xrow[k + khalf];
        a.y = xrow[k + khalf + 1];

        // B fragments: one ds_load_b64 each from the swizzled layout
        const v2f* __restrict__ wrow = sWp + ((k + khalf) >> 1) * D_OUT + m;
        v2f b0 = wrow[0];
        v2f b1 = wrow[16];
        v2f b2 = wrow[32];
        v2f b3 = wrow[48];

        acc0 = __builtin_amdgcn_wmma_f32_16x16x4_f32(false, a, false, b0, (short)0, acc0, false, false);
        acc1 = __builtin_amdgcn_wmma_f32_16x16x4_f32(false, a, false, b1, (short)0, acc1, false, false);
        acc2 = __builtin_amdgcn_wmma_f32_16x16x4_f32(false, a, false, b2, (short)0, acc2, false, false);
        acc3 = __builtin_amdgcn_wmma_f32_16x16x4_f32(false, a, false, b3, (short)0, acc3, false, false);
    }

    // C/D layout: VGPR v -> M = v (lanes 0-15) or v+8 (lanes 16-31); N = lane%16
    const int mbase = (lane >> 4) << 3;
    float* __restrict__ outT = P + (size_t)row0 * D_OUT;
    if (row0 + 16 <= nRows) {
#pragma unroll
        for (int v = 0; v < 8; ++v) {
            size_t o = (size_t)(mbase + v) * D_OUT + m;
            outT[o +  0] = acc0[v];
            outT[o + 16] = acc1[v];
            outT[o + 32] = acc2[v];
            outT[o + 48] = acc3[v];
        }
    } else {
#pragma unroll
        for (int v = 0; v < 8; ++v) {
            int r = mbase + v;
            if (row0 + r < nRows) {
                size_t o = (size_t)r * D_OUT + m;
                outT[o +  0] = acc0[v];
                outT[o + 16] = acc1[v];
                outT[o + 32] = acc2[v];
                outT[o + 48] = acc3[v];
            }
        }
    }
}

// ---------------------------------------------------------------------------
// K2: out[dst] += edge_val * pre_sup[src]  (COO scatter-add)
// One wave per edge; each lane handles 2 of the 64 columns (float2 gather,
// 2x f32 hardware atomics that resolve in L2 since `out` fits in 192MB L2).
// ---------------------------------------------------------------------------
__global__ __launch_bounds__(256) void gc_scatter_kernel(const float* __restrict__ P,
                                                         const float* __restrict__ edge_val,
                                                         const int* __restrict__ edge_src,
                                                         const int* __restrict__ edge_dst,
                                                         float* __restrict__ out,
                                                         int nEdges) {
    const int wave = threadIdx.x >> 5;
    const int lane = threadIdx.x & 31;
    const int e = blockIdx.x * 8 + wave;
    if (e >= nEdges) return;

    const int src = edge_src[e];
    const int dst = edge_dst[e];
    const float val = edge_val[e];

    const float2* __restrict__ ps = (const float2*)(P + (size_t)src * D_OUT);
    float2 v = ps[lane];

    float* __restrict__ o = out + (size_t)dst * D_OUT + lane * 2;
    unsafeAtomicAdd(o, v.x * val);
    unsafeAtomicAdd(o + 1, v.y * val);
}

// ---------------------------------------------------------------------------
// K3: per-column sum and sum-of-squares over [nRows x 64]
// threadIdx%64 = column, 4 row-lanes per block, register accumulation,
// one pair of atomics per thread at the end.
// ---------------------------------------------------------------------------
__global__ __launch_bounds__(256) void gc_stats_kernel(const float* __restrict__ out,
                                                       float* __restrict__ stats,
                                                       int nRows) {
    const int c = threadIdx.x & 63;
    const int rg = threadIdx.x >> 6;  // 0..3
    float s = 0.0f, s2 = 0.0f;
    for (int r = blockIdx.x * 4 + rg; r < nRows; r += gridDim.x * 4) {
        float v = out[(size_t)r * D_OUT + c];
        s += v;
        s2 += v * v;
    }
    unsafeAtomicAdd(&stats[c], s);
    unsafeAtomicAdd(&stats[D_OUT + c], s2);
}

// ---------------------------------------------------------------------------
// K4: fold sums into mean and rsqrt(var + eps) (64 threads)
// ---------------------------------------------------------------------------
__global__ void gc_finstats_kernel(float* __restrict__ stats, float invN) {
    int c = threadIdx.x;
    if (c < D_OUT) {
        float mean = stats[c] * invN;
        float var = stats[D_OUT + c] * invN - mean * mean;
        stats[c] = mean;
        stats[D_OUT + c] = rsqrtf(var + BN_EPS);
    }
}

// ---------------------------------------------------------------------------
// K5: out = relu((out - mean) * inv_std), in place
// ---------------------------------------------------------------------------
__global__ __launch_bounds__(256) void gc_norm_kernel(float* __restrict__ out,
                                                      const float* __restrict__ stats,
                                                      int total) {
    int i = blockIdx.x * 256 + threadIdx.x;
    if (i >= total) return;
    int c = i & 63;
    float v = (out[i] - stats[c]) * stats[D_OUT + c];
    out[i] = v > 0.0f ? v : 0.0f;
}

// ---------------------------------------------------------------------------
extern "C" void kernel_launch(void* const* d_in, const int* in_sizes, int n_in,
                              void* d_out, int out_size, void* d_ws, size_t ws_size,
                              hipStream_t stream) {
    const float* x        = (const float*)d_in[0];  // [N, 128]
    const float* W        = (const float*)d_in[1];  // [128, 64]
    const float* edge_val = (const float*)d_in[2];  // [E]
    const int*   edge_src = (const int*)d_in[3];    // [E]
    const int*   edge_dst = (const int*)d_in[4];    // [E]
    float* out = (float*)d_out;                     // [N, 64]

    const int nRows  = in_sizes[0] / D_IN;          // 100000
    const int nEdges = in_sizes[2];                 // 3200000
    const int total  = nRows * D_OUT;

    float* P     = (float*)d_ws;                    // pre_sup [N, 64]
    float* stats = P + (size_t)nRows * D_OUT;       // sum[64] | sumsq[64]

    gc_zero_kernel<<<(total + 255) / 256, 256, 0, stream>>>(out, total, stats);

    const int tiles = (nRows + 15) / 16;            // 16-row WMMA tiles
    gc_gemm_kernel<<<(tiles + 7) / 8, 256, 0, stream>>>(x, W, P, nRows);

    gc_scatter_kernel<<<(nEdges + 7) / 8, 256, 0, stream>>>(P, edge_val, edge_src,
                                                            edge_dst, out, nEdges);

    gc_stats_kernel<<<512, 256, 0, stream>>>(out, stats, nRows);
    gc_finstats_kernel<<<1, 64, 0, stream>>>(stats, 1.0f / (float)nRows);
    gc_norm_kernel<<<(total + 255) / 256, 256, 0, stream>>>(out, stats, total);
}